// RobustAttentionHead_20538533609918
// MI455X (gfx1250) — compile-verified
//
#include <hip/hip_runtime.h>
#include <hip/hip_bf16.h>

typedef __attribute__((ext_vector_type(2))) float v2f;
typedef __attribute__((ext_vector_type(8))) float v8f;

#define NNODES 50000
#define DIM 128
#define HEADS 8
#define CH 16
#define NEDGE 800000
#define ETOT (NEDGE + NNODES)
#define NEG_SLOPE 0.2f
#define LN_EPS 1e-5f

// ---------------------------------------------------------------------------
// Pass 0: zero the scratch accumulators (accum, denom, mord).
// mord == 0u is the minimum of the order-preserving float encoding, so it
// doubles as -inf for the segment max (every node has a self loop anyway).
// ---------------------------------------------------------------------------
__global__ __launch_bounds__(256)
void init_ws_kernel(float* __restrict__ accum, float* __restrict__ denom,
                    unsigned* __restrict__ mord) {
    size_t i = (size_t)blockIdx.x * blockDim.x + threadIdx.x;
    if (i < (size_t)NNODES * DIM) accum[i] = 0.0f;
    if (i < (size_t)NNODES * HEADS) { denom[i] = 0.0f; mord[i] = 0u; }
}

// ---------------------------------------------------------------------------
// Pass 1: xl = x@Wl + bl, xr = x@Wr + br via V_WMMA_F32_16X16X4_F32.
// Block = 256 threads = 8 waves; block handles 16 node rows; wave w owns
// output columns [16w, 16w+16) and keeps two f32 accumulators (Wl and Wr).
// K loop: 128/4 = 32 steps -> 64 WMMAs per wave.
//
// Fragment layouts (ISA 7.12.2, 32-bit, wave32):
//   A 16x4 : lane l -> M = l&15 ; VGPR v holds K = k0 + 2*(l>>4) + v
//   B 4x16 : lane l -> N = l&15 ; VGPR v holds K = k0 + 2*(l>>4) + v
//   C 16x16: lane l -> N = l&15 ; VGPR v holds M = v + 8*(l>>4)
// ---------------------------------------------------------------------------
__global__ __launch_bounds__(256)
void gatv2_linear_wmma_kernel(const float* __restrict__ x,
                              const float* __restrict__ Wl, const float* __restrict__ bl,
                              const float* __restrict__ Wr, const float* __restrict__ br,
                              float* __restrict__ xl, float* __restrict__ xr) {
    const int lane = threadIdx.x & 31;
    const int wave = threadIdx.x >> 5;
    const int half = lane >> 4;
    const int l15  = lane & 15;
    const int rowbase = blockIdx.x * 16;
    const int n = wave * 16 + l15;          // output column this lane owns (B/C/D)

    v8f accL = {};
    v8f accR = {};

    const float* arow = x + (size_t)(rowbase + l15) * DIM;   // A row for this lane
    for (int k0 = 0; k0 < DIM; k0 += 4) {
        const int kk = k0 + 2 * half;
        v2f a = *(const v2f*)(arow + kk);                    // A[m][kk], A[m][kk+1]
        v2f bL, bR;
        bL.x = Wl[(size_t)kk * DIM + n];
        bL.y = Wl[(size_t)(kk + 1) * DIM + n];
        bR.x = Wr[(size_t)kk * DIM + n];
        bR.y = Wr[(size_t)(kk + 1) * DIM + n];
        accL = __builtin_amdgcn_wmma_f32_16x16x4_f32(false, a, false, bL,
                                                     (short)0, accL, false, false);
        accR = __builtin_amdgcn_wmma_f32_16x16x4_f32(false, a, false, bR,
                                                     (short)0, accR, false, false);
    }

    const float bln = bl[n];
    const float brn = br[n];
#pragma unroll
    for (int v = 0; v < 8; ++v) {
        const int m = v + 8 * half;
        const size_t off = (size_t)(rowbase + m) * DIM + n;
        xl[off] = accL[v] + bln;
        xr[off] = accR[v] + brn;
    }
}

// order-preserving float <-> uint encoding for atomic segment-max
__device__ __forceinline__ unsigned ord_encode(float f) {
    unsigned u = __float_as_uint(f);
    return (u & 0x80000000u) ? ~u : (u | 0x80000000u);
}
__device__ __forceinline__ float ord_decode(unsigned u) {
    return (u & 0x80000000u) ? __uint_as_float(u ^ 0x80000000u)
                             : __uint_as_float(~u);
}

__device__ __forceinline__ void edge_src_dst(int e, const int* __restrict__ ei,
                                             int& src, int& dst) {
    if (e < NEDGE) { src = ei[e]; dst = ei[NEDGE + e]; }
    else           { src = e - NEDGE; dst = src; }        // self loops appended
}

// ---------------------------------------------------------------------------
// Pass 2: one thread per (edge, head): logit = att_h . leaky_relu(xl_src+xr_dst)
// store logit, atomic segment-max into mord[dst,h].
// ---------------------------------------------------------------------------
__global__ __launch_bounds__(256)
void edge_logits_kernel(const float* __restrict__ xl, const float* __restrict__ xr,
                        const int* __restrict__ ei, const float* __restrict__ att,
                        float* __restrict__ logits, unsigned* __restrict__ mord) {
    const long long tid = (long long)blockIdx.x * blockDim.x + threadIdx.x;
    if (tid >= (long long)ETOT * HEADS) return;
    const int e = (int)(tid >> 3);
    const int h = (int)(tid & 7);
    int src, dst;
    edge_src_dst(e, ei, src, dst);

    const float4* pa = (const float4*)(xl + (size_t)src * DIM + h * CH);
    const float4* pb = (const float4*)(xr + (size_t)dst * DIM + h * CH);
    const float4* pw = (const float4*)(att + h * CH);
    float acc = 0.0f;
#pragma unroll
    for (int i = 0; i < 4; ++i) {
        float4 a = pa[i], b = pb[i], w = pw[i];
        float s0 = a.x + b.x, s1 = a.y + b.y, s2 = a.z + b.z, s3 = a.w + b.w;
        s0 = (s0 >= 0.0f) ? s0 : NEG_SLOPE * s0;
        s1 = (s1 >= 0.0f) ? s1 : NEG_SLOPE * s1;
        s2 = (s2 >= 0.0f) ? s2 : NEG_SLOPE * s2;
        s3 = (s3 >= 0.0f) ? s3 : NEG_SLOPE * s3;
        acc = fmaf(s0, w.x, fmaf(s1, w.y, fmaf(s2, w.z, fmaf(s3, w.w, acc))));
    }
    logits[tid] = acc;
    atomicMax(&mord[(size_t)dst * HEADS + h], ord_encode(acc));
}

// ---------------------------------------------------------------------------
// Pass 3: a = exp(logit - max); atomicAdd denom[dst,h] += a;
// atomicAdd accum[dst, h*16+c] += a * xl[src, h*16+c]  (unnormalized messages;
// the 1/denom factor is applied once per node in the finalize pass).
// ---------------------------------------------------------------------------
__global__ __launch_bounds__(256)
void edge_accum_kernel(const float* __restrict__ xl, const int* __restrict__ ei,
                       const float* __restrict__ logits,
                       const unsigned* __restrict__ mord,
                       float* __restrict__ denom, float* __restrict__ accum) {
    const long long tid = (long long)blockIdx.x * blockDim.x + threadIdx.x;
    if (tid >= (long long)ETOT * HEADS) return;
    const int e = (int)(tid >> 3);
    const int h = (int)(tid & 7);
    int src, dst;
    edge_src_dst(e, ei, src, dst);

    const float m = ord_decode(mord[(size_t)dst * HEADS + h]);
    const float a = __expf(logits[tid] - m);
    atomicAdd(&denom[(size_t)dst * HEADS + h], a);

    const float4* ps = (const float4*)(xl + (size_t)src * DIM + h * CH);
    float* pd = accum + (size_t)dst * DIM + h * CH;
#pragma unroll
    for (int i = 0; i < 4; ++i) {
        float4 v = ps[i];
        atomicAdd(&pd[4 * i + 0], a * v.x);
        atomicAdd(&pd[4 * i + 1], a * v.y);
        atomicAdd(&pd[4 * i + 2], a * v.z);
        atomicAdd(&pd[4 * i + 3], a * v.w);
    }
}

// ---------------------------------------------------------------------------
// Pass 4: one wave32 per node, 4 channels per lane.
// out = accum/denom + bias ; y = x + out ; LayerNorm(y) -> d_out.
// Mean/var reductions via __shfl_xor across the wave (no LDS needed).
// ---------------------------------------------------------------------------
__global__ __launch_bounds__(256)
void node_finalize_kernel(const float* __restrict__ x, const float* __restrict__ accum,
                          const float* __restrict__ denom, const float* __restrict__ bias,
                          const float* __restrict__ gamma, const float* __restrict__ beta,
                          float* __restrict__ out) {
    const int lane = threadIdx.x & 31;
    const int node = blockIdx.x * 8 + (threadIdx.x >> 5);
    if (node >= NNODES) return;
    const int c0 = lane * 4;              // 4 contiguous channels, never crosses head
    const int h  = lane >> 2;             // head index = c0 / 16

    const float inv_d = 1.0f / denom[(size_t)node * HEADS + h];
    float4 ac = *(const float4*)(accum + (size_t)node * DIM + c0);
    float4 xv = *(const float4*)(x + (size_t)node * DIM + c0);
    float4 bv = *(const float4*)(bias + c0);

    float4 y;
    y.x = xv.x + ac.x * inv_d + bv.x;
    y.y = xv.y + ac.y * inv_d + bv.y;
    y.z = xv.z + ac.z * inv_d + bv.z;
    y.w = xv.w + ac.w * inv_d + bv.w;

    float s = y.x + y.y + y.z + y.w;
#pragma unroll
    for (int m = 16; m >= 1; m >>= 1) s += __shfl_xor(s, m, 32);
    const float mu = s * (1.0f / DIM);

    float dx = y.x - mu, dy = y.y - mu, dz = y.z - mu, dw = y.w - mu;
    float q = dx * dx + dy * dy + dz * dz + dw * dw;
#pragma unroll
    for (int m = 16; m >= 1; m >>= 1) q += __shfl_xor(q, m, 32);
    const float rstd = rsqrtf(q * (1.0f / DIM) + LN_EPS);

    float4 g  = *(const float4*)(gamma + c0);
    float4 bt = *(const float4*)(beta + c0);
    float4 o;
    o.x = dx * rstd * g.x + bt.x;
    o.y = dy * rstd * g.y + bt.y;
    o.z = dz * rstd * g.z + bt.z;
    o.w = dw * rstd * g.w + bt.w;
    *(float4*)(out + (size_t)node * DIM + c0) = o;
}

extern "C" void kernel_launch(void* const* d_in, const int* in_sizes, int n_in,
                              void* d_out, int out_size, void* d_ws, size_t ws_size,
                              hipStream_t stream) {
    const float* x     = (const float*)d_in[0];
    const int*   ei    = (const int*)d_in[1];   // (2, E) int32, row-major
    const float* Wl    = (const float*)d_in[2];
    const float* bl    = (const float*)d_in[3];
    const float* Wr    = (const float*)d_in[4];
    const float* br    = (const float*)d_in[5];
    const float* att   = (const float*)d_in[6];
    const float* bias  = (const float*)d_in[7];
    const float* gamma = (const float*)d_in[8];
    const float* beta  = (const float*)d_in[9];
    float* out = (float*)d_out;

    // Workspace layout (floats):
    //   xl[N*128] | xr[N*128] | accum[N*128] | denom[N*8] | mord[N*8] | logits[ETOT*8]
    float* xl    = (float*)d_ws;
    float* xr    = xl + (size_t)NNODES * DIM;
    float* accum = xr + (size_t)NNODES * DIM;
    float* denom = accum + (size_t)NNODES * DIM;
    unsigned* mord = (unsigned*)(denom + (size_t)NNODES * HEADS);
    float* logits  = (float*)(mord + (size_t)NNODES * HEADS);

    // Pass 0: init accumulators (workspace is poisoned by the harness)
    {
        const size_t n = (size_t)NNODES * DIM;
        const int blocks = (int)((n + 255) / 256);
        init_ws_kernel<<<blocks, 256, 0, stream>>>(accum, denom, mord);
    }
    // Pass 1: WMMA dual projection
    {
        const int blocks = NNODES / 16;   // 50000 divides evenly into 16-row tiles
        gatv2_linear_wmma_kernel<<<blocks, 256, 0, stream>>>(x, Wl, bl, Wr, br, xl, xr);
    }
    // Pass 2: edge logits + segment max
    {
        const long long n = (long long)ETOT * HEADS;
        const int blocks = (int)((n + 255) / 256);
        edge_logits_kernel<<<blocks, 256, 0, stream>>>(xl, xr, ei, att, logits, mord);
    }
    // Pass 3: exp + denom + unnormalized message aggregation
    {
        const long long n = (long long)ETOT * HEADS;
        const int blocks = (int)((n + 255) / 256);
        edge_accum_kernel<<<blocks, 256, 0, stream>>>(xl, ei, logits, mord, denom, accum);
    }
    // Pass 4: normalize + bias + residual + LayerNorm
    {
        const int blocks = (NNODES + 7) / 8;
        node_finalize_kernel<<<blocks, 256, 0, stream>>>(x, accum, denom, bias,
                                                         gamma, beta, out);
    }
}